// FillDiagonalsFromArray2_25417616458410
// MI455X (gfx1250) — compile-verified
//
#include <hip/hip_runtime.h>
#include <stdint.h>

// out[n, i, j] = x[n, |i - j|]   (N=64, m=1024 in the reference setup)
//
// Store-bandwidth-bound: 256 MB out @ 23.3 TB/s ~= 11 us floor.
//   - stage x[n, :] (4 KB) into LDS once per block via gfx1250 async
//     global->LDS b128 (ASYNCcnt path), completed with s_wait_asynccnt
//   - Toeplitz recurrence: out[i+1][j] == out[i][j-1], so after the first
//     row each thread shifts its 4-vector and gathers ONE new LDS element
//   - non-temporal b128 stores (write-once stream; 256 MB > 192 MB L2)

typedef float v4f __attribute__((ext_vector_type(4)));

#define BLOCK  256
#define TILE_I 16

__device__ __forceinline__ void async_stage_row(const float* __restrict__ gsrc,
                                                float* s_row, int m, int t)
{
    for (int off = t * 4; off < m; off += BLOCK * 4) {
        unsigned lds_b = (unsigned)(uintptr_t)(&s_row[off]); // low 32 bits = LDS byte offset
        const float* g = gsrc + off;
        asm volatile("global_load_async_to_lds_b128 %0, %1, off"
                     :
                     : "v"(lds_b), "v"(g)
                     : "memory");
    }
    asm volatile("s_wait_asynccnt 0" ::: "memory");
}

// ---- fast path: m known at compile time (1024) ----------------------------
template <int M>
__global__ void __launch_bounds__(BLOCK)
toeplitz_fixed_kernel(const float* __restrict__ x, float* __restrict__ out)
{
    __shared__ float s_row[M];

    const int t = threadIdx.x;
    const int n = blockIdx.y;

    async_stage_row(x + (long long)n * M, s_row, M, t);
    __syncthreads();

    const int i0 = blockIdx.x * TILE_I;
    const int j0 = t * 4;
    float* base = out + ((long long)n * M + i0) * M + j0;

    int d = i0 - j0;                 // i - j for element .x of row r=0
    v4f v;
    v.x = s_row[abs(d)];
    v.y = s_row[abs(d - 1)];
    v.z = s_row[abs(d - 2)];
    v.w = s_row[abs(d - 3)];
    __builtin_nontemporal_store(v, (v4f*)base);

    #pragma unroll
    for (int r = 1; r < TILE_I; ++r) {
        d += 1;
        // out[i+1][j] = out[i][j-1]: shift within the 4-chunk, gather one.
        v.w = v.z;
        v.z = v.y;
        v.y = v.x;
        v.x = s_row[abs(d)];
        __builtin_nontemporal_store(v, (v4f*)(base + r * M)); // imm offset r*4096B
    }
}

// ---- generic fallback (any m, multiple of 4) ------------------------------
__global__ void __launch_bounds__(BLOCK)
toeplitz_generic_kernel(const float* __restrict__ x, float* __restrict__ out, int m)
{
    extern __shared__ float s_row[];

    const int t = threadIdx.x;
    const int n = blockIdx.y;

    async_stage_row(x + (long long)n * m, s_row, m, t);
    __syncthreads();

    const int i0 = blockIdx.x * TILE_I;
    float* out_n = out + (long long)n * m * m;

    for (int r = 0; r < TILE_I; ++r) {
        const int i = i0 + r;
        if (i >= m) break;
        float* orow = out_n + (long long)i * m;
        for (int j = t * 4; j < m; j += BLOCK * 4) {
            const int d0 = i - j;
            v4f v;
            v.x = s_row[abs(d0)];
            v.y = s_row[abs(d0 - 1)];
            v.z = s_row[abs(d0 - 2)];
            v.w = s_row[abs(d0 - 3)];
            __builtin_nontemporal_store(v, (v4f*)(orow + j));
        }
    }
}

extern "C" void kernel_launch(void* const* d_in, const int* in_sizes, int n_in,
                              void* d_out, int out_size, void* d_ws, size_t ws_size,
                              hipStream_t stream)
{
    (void)n_in; (void)d_ws; (void)ws_size;

    const float* x = (const float*)d_in[0];
    float* out = (float*)d_out;

    const long long in_elems = (long long)in_sizes[0];   // N * m
    if (in_elems <= 0 || out_size <= 0) return;

    const int m = (int)((long long)out_size / in_elems); // out = N*m*m
    const int N = (int)(in_elems / m);

    dim3 grid((m + TILE_I - 1) / TILE_I, N);

    if (m == 1024) {
        toeplitz_fixed_kernel<1024><<<grid, BLOCK, 0, stream>>>(x, out);
    } else {
        const size_t shmem = (size_t)m * sizeof(float);
        toeplitz_generic_kernel<<<grid, BLOCK, shmem, stream>>>(x, out, m);
    }
}